// SmearImages_64046552318460
// MI455X (gfx1250) — compile-verified
//
#include <hip/hip_runtime.h>
#include <math.h>

typedef float v2f __attribute__((ext_vector_type(2)));
typedef float v8f __attribute__((ext_vector_type(8)));

#define GRID_N   (64 * 64 * 64)   // 262144 points
#define NIMG     16
#define NCH      3
#define IMH      480
#define IMW      640
#define IMHW     (IMH * IMW)
#define PTS_BLK  128              // points per block
#define THREADS  256              // 8 waves (wave32)
#define PSTRIDE  (PTS_BLK + 2)    // LDS row pad to avoid bank conflicts

__global__ __launch_bounds__(THREADS)
void SmearImages_kernel(const float* __restrict__ coord,   // (3, N)
                        const float* __restrict__ images,  // (I, C, H, W)
                        const float* __restrict__ Tm,      // (I, 3, 4) = K @ T_cw[:3]
                        const float* __restrict__ Tcw,     // (I, 4, 4)
                        float* __restrict__ out)           // (I, 8, N)
{
    __shared__ float proj_lds[48 * PSTRIDE];       // proj rows 0..47 x 128 points
    __shared__ float pts_lds[3 * PTS_BLK];         // staged coordinates
    __shared__ float centers_lds[NIMG * 3];        // camera centers -R^T t

    const int tid  = threadIdx.x;
    const int lane = tid & 31;
    const int wave = tid >> 5;
    const int base = blockIdx.x * PTS_BLK;

    // ---- stage point coords into LDS (coalesced) ----
    for (int k = tid; k < 3 * PTS_BLK; k += THREADS) {
        int d = k >> 7;           // dimension 0..2
        int p = k & 127;          // local point
        pts_lds[k] = coord[d * GRID_N + base + p];
    }
    // ---- camera centers: c_j = -sum_k R[k][j] * t[k] ----
    if (tid < NIMG * 3) {
        int i = tid / 3, j = tid - 3 * i;
        const float* T = Tcw + i * 16;
        centers_lds[tid] = -(T[0 * 4 + j] * T[0 * 4 + 3] +
                             T[1 * 4 + j] * T[1 * 4 + 3] +
                             T[2 * 4 + j] * T[2 * 4 + 3]);
    }

    // ---- Phase 1: projection GEMM via V_WMMA_F32_16X16X4_F32 ----
    // B (4x16): homogeneous coords of this wave's 16 points.
    //   lanes 0-15 : VGPR0=K0(x), VGPR1=K1(y);  lanes 16-31 : VGPR0=K2(z), VGPR1=K3(1)
    const int n0  = base + wave * 16;
    const int col = lane & 15;
    v2f b;
    if (lane < 16) {
        b.x = coord[0 * GRID_N + n0 + col];
        b.y = coord[1 * GRID_N + n0 + col];
    } else {
        b.x = coord[2 * GRID_N + n0 + col];
        b.y = 1.0f;
    }
    const int kh = (lane >> 4) * 2;   // K-half: 0 or 2
    #pragma unroll
    for (int blk = 0; blk < 3; ++blk) {
        // A (16x4): rows blk*16 .. blk*16+15 of the stacked 48x4 transform matrix
        int row = blk * 16 + col;
        v2f a;
        a.x = Tm[row * 4 + kh];
        a.y = Tm[row * 4 + kh + 1];
        v8f c = {};
        c = __builtin_amdgcn_wmma_f32_16x16x4_f32(
                /*neg_a=*/false, a, /*neg_b=*/false, b,
                /*c_mod=*/(short)0, c, /*reuse_a=*/false, /*reuse_b=*/false);
        // C layout: VGPR r, lane l -> (M = r + 8*(l>=16), N = l&15)
        int mbase = blk * 16 + ((lane >> 4) << 3);
        #pragma unroll
        for (int r = 0; r < 8; ++r)
            proj_lds[(mbase + r) * PSTRIDE + wave * 16 + col] = c[r];
    }
    __syncthreads();

    // ---- Phase 2: divide / gather / view-dir / store (2048 pairs per block) ----
    #pragma unroll
    for (int c8 = 0; c8 < 8; ++c8) {
        int idx = c8 * THREADS + tid;   // 0..2047
        int i   = idx >> 7;             // image (uniform per wave)
        int nl  = idx & 127;            // local point (consecutive per lane)
        int n   = base + nl;

        float px = proj_lds[(3 * i + 0) * PSTRIDE + nl];
        float py = proj_lds[(3 * i + 1) * PSTRIDE + nl];
        float pz = proj_lds[(3 * i + 2) * PSTRIDE + nl];
        float denom = pz + 1e-8f;
        float u = px / denom;
        float v = py / denom;
        bool valid = (u >= 0.0f) && (u <= (float)(IMW - 1)) &&
                     (v >= 0.0f) && (v <= (float)(IMH - 1)) && (pz > 0.0f);
        float validf = valid ? 1.0f : 0.0f;
        float ur = fminf(fmaxf(rintf(u), 0.0f), (float)(IMW - 1));
        float vr = fminf(fmaxf(rintf(v), 0.0f), (float)(IMH - 1));
        int flat = (int)vr * IMW + (int)ur;

        const float* img = images + (size_t)i * (NCH * IMHW) + flat;
        float f0 = img[0 * IMHW] * validf;
        float f1 = img[1 * IMHW] * validf;
        float f2 = img[2 * IMHW] * validf;

        float dx = pts_lds[0 * PTS_BLK + nl] - centers_lds[i * 3 + 0];
        float dy = pts_lds[1 * PTS_BLK + nl] - centers_lds[i * 3 + 1];
        float dz = pts_lds[2 * PTS_BLK + nl] - centers_lds[i * 3 + 2];
        float inv = 1.0f / (sqrtf(dx * dx + dy * dy + dz * dz) + 1e-8f);

        float* o = out + (size_t)i * 8 * GRID_N + n;
        o[0 * GRID_N] = f0;
        o[1 * GRID_N] = f1;
        o[2 * GRID_N] = f2;
        o[3 * GRID_N] = pz;        // raw depth (written even when invalid)
        o[4 * GRID_N] = validf;
        o[5 * GRID_N] = dx * inv;
        o[6 * GRID_N] = dy * inv;
        o[7 * GRID_N] = dz * inv;
    }
}

extern "C" void kernel_launch(void* const* d_in, const int* in_sizes, int n_in,
                              void* d_out, int out_size, void* d_ws, size_t ws_size,
                              hipStream_t stream) {
    const float* coord  = (const float*)d_in[0];  // (3, 64, 64, 64)
    const float* images = (const float*)d_in[1];  // (16, 3, 480, 640)
    const float* Tm     = (const float*)d_in[2];  // (16, 3, 4)
    const float* Tcw    = (const float*)d_in[3];  // (16, 4, 4)
    float* out = (float*)d_out;                   // (16, 8, 64, 64, 64)

    dim3 grid(GRID_N / PTS_BLK);   // 2048 blocks
    dim3 block(THREADS);           // 256 threads = 8 wave32
    SmearImages_kernel<<<grid, block, 0, stream>>>(coord, images, Tm, Tcw, out);
}